// CombinedLoss_54631984005443
// MI455X (gfx1250) — compile-verified
//
#include <hip/hip_runtime.h>
#include <math.h>

typedef __attribute__((ext_vector_type(2))) float v2f;
typedef __attribute__((ext_vector_type(8))) float v8f;

#define INFBITS 0x7F800000u

// Raw v_sqrt_f32 (~1 ulp, no IEEE fixup sequence) — inputs are clamped >= 0.
__device__ __forceinline__ float fast_sqrtf(float x) {
  return __builtin_amdgcn_sqrtf(x);
}

// ---------------------------------------------------------------------------
// Workspace init: min-buffers -> +inf bits, EMD accumulator -> 0.0f
// ---------------------------------------------------------------------------
__global__ void init_ws_kernel(unsigned* __restrict__ ws, int nMin) {
  int i = blockIdx.x * blockDim.x + threadIdx.x;
  if (i < nMin) ws[i] = INFBITS;
  else if (i == nMin) ws[i] = 0u;  // emd accumulator = 0.0f
}

// ---------------------------------------------------------------------------
// Per-wave 16x128 strip of a 128x128 distance tile via WMMA f32 16x16x4.
// FULL=true: tile fully in-range, no bounds selects in the epilogue.
// ---------------------------------------------------------------------------
template <bool FULL>
__device__ __forceinline__ void dist_tile_body(
    const float* Pl, const float* Ql, const float* Pn, const float* Qn,
    unsigned* rmS, unsigned* cmS,
    int rowBase, int colBase, int nP, int nQ,
    bool wantRow, bool wantCol, int wave, int lane) {
  const float FINF = __uint_as_float(INFBITS);
  int lh = lane >> 4;      // half-wave: K=0,1 (lanes 0-15) vs K=2,3 (16-31)
  int ll = lane & 15;

  // A fragment: rows [wave*16, wave*16+16); one branchless ds_load_b64/lane
  int pr = wave * 16 + ll;
  v2f a = *(const v2f*)&Pl[pr * 4 + 2 * lh];

  // Row norms for this wave's strip are sub-tile invariant: hoist to regs.
  float na[8];
#pragma unroll
  for (int r = 0; r < 8; r++) na[r] = Pn[wave * 16 + r + 8 * lh];

  float rm[8];
#pragma unroll
  for (int r = 0; r < 8; r++) rm[r] = FINF;

#pragma unroll
  for (int t = 0; t < 8; t++) {
    int qc = t * 16 + ll;
    v2f bf = *(const v2f*)&Ql[qc * 4 + 2 * lh];  // one ds_load_b64/lane

    v8f acc = {};
    // D(16x16) = A(16x4) * B(4x16): dot(p_m, q_n) for the whole sub-tile
    acc = __builtin_amdgcn_wmma_f32_16x16x4_f32(
        false, a, false, bf, (short)0, acc, false, false);

    float nb = Qn[qc];
    bool cv = FULL || (colBase + qc) < nQ;
    float cmv = FINF;
#pragma unroll
    for (int r = 0; r < 8; r++) {
      // C/D layout: element (M = r + 8*(lane>=16), N = lane%16)
      float d2 = na[r] + nb - 2.0f * acc[r];
      float d = fast_sqrtf(fmaxf(d2, 0.f));
      if (FULL) {
        rm[r] = fminf(rm[r], d);
        cmv = fminf(cmv, d);
      } else {
        bool rv = (rowBase + wave * 16 + r + 8 * lh) < nP;
        rm[r] = fminf(rm[r], cv ? d : FINF);
        cmv = fminf(cmv, rv ? d : FINF);
      }
    }
    if (wantCol) {
      cmv = fminf(cmv, __shfl_xor(cmv, 16, 32));  // merge M-halves of column
      if (lh == 0) atomicMin(&cmS[qc], __float_as_uint(cmv));
    }
  }

  if (wantRow) {
#pragma unroll
    for (int r = 0; r < 8; r++) {
      float v = rm[r];
      v = fminf(v, __shfl_xor(v, 8, 32));   // reduce over 16 N-lanes in half
      v = fminf(v, __shfl_xor(v, 4, 32));
      v = fminf(v, __shfl_xor(v, 2, 32));
      v = fminf(v, __shfl_xor(v, 1, 32));
      if (ll == 0) atomicMin(&rmS[wave * 16 + r + 8 * lh], __float_as_uint(v));
    }
  }
}

// ---------------------------------------------------------------------------
// Pairwise-distance tile kernel with WMMA f32 16x16x4 (K=3 padded to 4).
// Block = 256 threads (8 waves), computes a 128x128 tile of D = dist(P, Q).
// Points are stored in LDS as [x,y,z,0] so each lane's A/B fragment is one
// branchless ds_load_b64. Row/col minima reduced via lane shuffles + LDS u32
// atomics (dist >= 0 so float bits are order-preserving as unsigned), then
// one global atomic min per row/col per block.
// ---------------------------------------------------------------------------
__global__ __launch_bounds__(256) void dist_min_kernel(
    const float* __restrict__ P, const float* __restrict__ Q,
    int nP, int nQ, int bsP, int bsQ,
    unsigned* __restrict__ rowMin, unsigned* __restrict__ colMin, int rmStride) {
  __shared__ float Pl[128 * 4];   // x,y,z,0 per point (WMMA K-pad)
  __shared__ float Ql[128 * 4];
  __shared__ float Pn[128];       // squared norms
  __shared__ float Qn[128];
  __shared__ unsigned rmS[128];   // per-tile row minima (u32-ordered float)
  __shared__ unsigned cmS[128];   // per-tile col minima

  int b = blockIdx.z;
  P += (size_t)b * bsP * 3;
  Q += (size_t)b * bsQ * 3;
  int rowBase = blockIdx.y * 128;
  int colBase = blockIdx.x * 128;
  int tid = threadIdx.x;

  if (tid < 128) {
    int r = rowBase + tid;
    float x = 0.f, y = 0.f, z = 0.f;
    if (r < nP) { x = P[r * 3]; y = P[r * 3 + 1]; z = P[r * 3 + 2]; }
    Pl[tid * 4 + 0] = x; Pl[tid * 4 + 1] = y; Pl[tid * 4 + 2] = z; Pl[tid * 4 + 3] = 0.f;
    Pn[tid] = x * x + y * y + z * z;
    rmS[tid] = INFBITS;
    cmS[tid] = INFBITS;
  } else {
    int t = tid - 128;
    int c = colBase + t;
    float x = 0.f, y = 0.f, z = 0.f;
    if (c < nQ) { x = Q[c * 3]; y = Q[c * 3 + 1]; z = Q[c * 3 + 2]; }
    Ql[t * 4 + 0] = x; Ql[t * 4 + 1] = y; Ql[t * 4 + 2] = z; Ql[t * 4 + 3] = 0.f;
    Qn[t] = x * x + y * y + z * z;
  }
  __syncthreads();

  int wave = tid >> 5;
  int lane = tid & 31;
  bool wantRow = rowMin != nullptr;
  bool wantCol = colMin != nullptr;

  if ((rowBase + 128 <= nP) && (colBase + 128 <= nQ)) {
    dist_tile_body<true>(Pl, Ql, Pn, Qn, rmS, cmS, rowBase, colBase, nP, nQ,
                         wantRow, wantCol, wave, lane);
  } else {
    dist_tile_body<false>(Pl, Ql, Pn, Qn, rmS, cmS, rowBase, colBase, nP, nQ,
                          wantRow, wantCol, wave, lane);
  }
  __syncthreads();

  if (tid < 128) {
    if (wantRow && (rowBase + tid) < nP)
      atomicMin(&rowMin[(size_t)b * rmStride + rowBase + tid], rmS[tid]);
    if (wantCol && (colBase + tid) < nQ)
      atomicMin(&colMin[colBase + tid], cmS[tid]);
  }
}

// ---------------------------------------------------------------------------
// Sinkhorn EMD: one 1024-thread workgroup per batch. Points + f/g live in
// LDS (40 KB << 320 KB/WGP); the 1024x1024 cost matrix is recomputed on the
// fly each pass (pure VALU, zero memory traffic in the 100-iteration loop).
// ---------------------------------------------------------------------------
__global__ __launch_bounds__(1024) void sinkhorn_kernel(
    const float* __restrict__ P, const float* __restrict__ Q,
    float* __restrict__ emdOut) {
  const int N = 1024;
  __shared__ float4 Ps[N];
  __shared__ float4 Qs[N];
  __shared__ float fv[N];
  __shared__ float gv[N];
  __shared__ float red[32];
  __shared__ float epsS[2];

  int b = blockIdx.x;
  const float* Pb = P + (size_t)b * N * 3;
  const float* Qb = Q + (size_t)b * N * 3;
  int tid = threadIdx.x;
  int wave = tid >> 5;
  int lane = tid & 31;

  {
    float4 p; p.x = Pb[tid * 3]; p.y = Pb[tid * 3 + 1]; p.z = Pb[tid * 3 + 2]; p.w = 0.f;
    Ps[tid] = p;
    float4 q; q.x = Qb[tid * 3]; q.y = Qb[tid * 3 + 1]; q.z = Qb[tid * 3 + 2]; q.w = 0.f;
    Qs[tid] = q;
    fv[tid] = 0.f;
    gv[tid] = 0.f;
  }
  __syncthreads();

  // eps = 0.02 * mean(C): pre-pass sum of all distances
  float sum = 0.f;
  for (int k = 0; k < 32; k++) {
    int i = wave * 32 + k;
    float4 p = Ps[i];
    for (int jb = 0; jb < 32; jb++) {
      int j = jb * 32 + lane;
      float4 q = Qs[j];
      float dx = p.x - q.x, dy = p.y - q.y, dz = p.z - q.z;
      sum += fast_sqrtf(dx * dx + dy * dy + dz * dz);
    }
  }
  for (int off = 16; off; off >>= 1) sum += __shfl_xor(sum, off, 32);
  if (lane == 0) red[wave] = sum;
  __syncthreads();
  if (tid == 0) {
    float t = 0.f;
    for (int w = 0; w < 32; w++) t += red[w];
    float eps = 0.02f * t / (float)(N * N);
    epsS[0] = eps;
    epsS[1] = 1.0f / eps;
  }
  __syncthreads();
  const float eps = epsS[0];
  const float inv_eps = epsS[1];
  const float elm = eps * (-__logf((float)N));  // eps * log_mu
  const float NEGINF = -__builtin_inff();

  for (int it = 0; it < 100; it++) {
    // f-pass: f_i = elm - eps * LSE_j((g_j - C_ij)/eps); wave owns 32 rows
    for (int k = 0; k < 32; k++) {
      int i = wave * 32 + k;
      float4 p = Ps[i];
      float m = NEGINF, s = 0.f;
      for (int jb = 0; jb < 32; jb++) {
        int j = jb * 32 + lane;
        float4 q = Qs[j];
        float dx = p.x - q.x, dy = p.y - q.y, dz = p.z - q.z;
        float d = fast_sqrtf(dx * dx + dy * dy + dz * dz);
        float v = (gv[j] - d) * inv_eps;
        float nm = fmaxf(m, v);
        s = s * __expf(m - nm) + __expf(v - nm);
        m = nm;
      }
      for (int off = 16; off; off >>= 1) {  // merge (m,s) LSE state across lanes
        float m2 = __shfl_xor(m, off, 32);
        float s2 = __shfl_xor(s, off, 32);
        float nm = fmaxf(m, m2);
        s = s * __expf(m - nm) + s2 * __expf(m2 - nm);
        m = nm;
      }
      if (lane == 0) fv[i] = elm - eps * (m + __logf(s));
    }
    __syncthreads();
    // g-pass: g_j = elm - eps * LSE_i((f_i - C_ij)/eps); wave owns 32 cols
    for (int k = 0; k < 32; k++) {
      int j = wave * 32 + k;
      float4 q = Qs[j];
      float m = NEGINF, s = 0.f;
      for (int ib = 0; ib < 32; ib++) {
        int i = ib * 32 + lane;
        float4 p = Ps[i];
        float dx = p.x - q.x, dy = p.y - q.y, dz = p.z - q.z;
        float d = fast_sqrtf(dx * dx + dy * dy + dz * dz);
        float v = (fv[i] - d) * inv_eps;
        float nm = fmaxf(m, v);
        s = s * __expf(m - nm) + __expf(v - nm);
        m = nm;
      }
      for (int off = 16; off; off >>= 1) {
        float m2 = __shfl_xor(m, off, 32);
        float s2 = __shfl_xor(s, off, 32);
        float nm = fmaxf(m, m2);
        s = s * __expf(m - nm) + s2 * __expf(m2 - nm);
        m = nm;
      }
      if (lane == 0) gv[j] = elm - eps * (m + __logf(s));
    }
    __syncthreads();
  }

  // sum(P * C) with P = exp((f_i + g_j - C_ij)/eps)
  float tot = 0.f;
  for (int k = 0; k < 32; k++) {
    int i = wave * 32 + k;
    float4 p = Ps[i];
    float fi = fv[i];
    for (int jb = 0; jb < 32; jb++) {
      int j = jb * 32 + lane;
      float4 q = Qs[j];
      float dx = p.x - q.x, dy = p.y - q.y, dz = p.z - q.z;
      float d = fast_sqrtf(dx * dx + dy * dy + dz * dz);
      tot += __expf((fi + gv[j] - d) * inv_eps) * d;
    }
  }
  for (int off = 16; off; off >>= 1) tot += __shfl_xor(tot, off, 32);
  if (lane == 0) red[wave] = tot;
  __syncthreads();
  if (tid == 0) {
    float t = 0.f;
    for (int w = 0; w < 32; w++) t += red[w];
    atomicAdd(emdOut, t);
  }
}

// ---------------------------------------------------------------------------
// Final scalar combine
// ---------------------------------------------------------------------------
__device__ __forceinline__ float block_sum256(float v, float* scratch, int tid) {
  int lane = tid & 31, wave = tid >> 5;
  for (int off = 16; off; off >>= 1) v += __shfl_xor(v, off, 32);
  if (lane == 0) scratch[wave] = v;
  __syncthreads();
  float r = 0.f;
  if (tid == 0)
    for (int w = 0; w < 8; w++) r += scratch[w];
  __syncthreads();
  return r;  // valid on tid 0
}

__global__ __launch_bounds__(256) void finalize_kernel(
    const unsigned* __restrict__ rowMin1, const unsigned* __restrict__ colMin1,
    const unsigned* __restrict__ rowMinS, const unsigned* __restrict__ colMinS,
    const unsigned* __restrict__ gtMin, const float* __restrict__ emdSum,
    const float* __restrict__ pc1_3, float* __restrict__ out,
    int nAll, int nSeed) {
  __shared__ float scratch[8];
  int tid = threadIdx.x;

  float sR1 = 0.f, sC1 = 0.f, sCS = 0.f, sConf = 0.f, sRS = 0.f;
  for (int i = tid; i < nAll; i += 256) {
    sR1 += __uint_as_float(rowMin1[i]);
    sC1 += __uint_as_float(colMin1[i]);
    sCS += __uint_as_float(colMinS[i]);
    float gt = __expf(-__uint_as_float(gtMin[i]));
    float dlt = pc1_3[i] - gt;
    sConf += dlt * dlt;
  }
  for (int i = tid; i < nSeed; i += 256) sRS += __uint_as_float(rowMinS[i]);

  float tR1 = block_sum256(sR1, scratch, tid);
  float tC1 = block_sum256(sC1, scratch, tid);
  float tCS = block_sum256(sCS, scratch, tid);
  float tCf = block_sum256(sConf, scratch, tid);
  float tRS = block_sum256(sRS, scratch, tid);

  if (tid == 0) {
    float cd = tC1 / (float)nAll + tR1 / (float)nAll;      // chamfer(pc1_0,pc2)
    float seed = tCS / (float)nAll + tRS / (float)nSeed;   // chamfer(pc1_1,pc2)
    float conf = tCf / (float)nAll;
    out[0] = conf + 0.5f * cd + 0.5f * emdSum[0] + seed;
  }
}

// ---------------------------------------------------------------------------
extern "C" void kernel_launch(void* const* d_in, const int* in_sizes, int n_in,
                              void* d_out, int out_size, void* d_ws, size_t ws_size,
                              hipStream_t stream) {
  const float* pc1_0 = (const float*)d_in[0];  // (8,1024,3)
  const float* pc1_1 = (const float*)d_in[1];  // (8,512,3)
  const float* pc1_3 = (const float*)d_in[2];  // (8,1024,1)
  const float* pc2   = (const float*)d_in[3];  // (8,1024,3)
  const float* pc3   = (const float*)d_in[4];  // (8,1024,3)

  const int B = 8;
  const int nAll = in_sizes[0] / 3;   // 8192 flattened pc1_0 / pc2 points
  const int nSeed = in_sizes[1] / 3;  // 4096 flattened pc1_1 points
  const int N = nAll / B;             // 1024 points per cloud

  // Workspace layout (u32 words)
  unsigned* ws = (unsigned*)d_ws;
  unsigned* rowMin1 = ws;                          // nAll  : dist2 (per pc1_0 pt)
  unsigned* colMin1 = rowMin1 + nAll;              // nAll  : dist1 (per pc2 pt)
  unsigned* rowMinS = colMin1 + nAll;              // nSeed : seed dist2
  unsigned* colMinS = rowMinS + nSeed;             // nAll  : seed dist1
  unsigned* gtMin   = colMinS + nAll;              // nAll  : per-batch pc3->pc2 min
  float*    emdSum  = (float*)(gtMin + nAll);      // 1     : EMD accumulator
  const int nMin = 4 * nAll + nSeed;               // words of +inf to write

  init_ws_kernel<<<(nMin + 1 + 255) / 256, 256, 0, stream>>>(ws, nMin);

  // Chamfer(pc1_0, pc2), flattened across batch: 8192 x 8192
  dist_min_kernel<<<dim3((nAll + 127) / 128, (nAll + 127) / 128, 1), 256, 0, stream>>>(
      pc1_0, pc2, nAll, nAll, 0, 0, rowMin1, colMin1, 0);

  // Seed chamfer(pc1_1, pc2): 4096 x 8192
  dist_min_kernel<<<dim3((nAll + 127) / 128, (nSeed + 127) / 128, 1), 256, 0, stream>>>(
      pc1_1, pc2, nSeed, nAll, 0, 0, rowMinS, colMinS, 0);

  // gt_scores: per-batch row mins of dist(pc3[b], pc2[b]), 1024 x 1024 x 8
  dist_min_kernel<<<dim3((N + 127) / 128, (N + 127) / 128, B), 256, 0, stream>>>(
      pc3, pc2, N, N, N, N, gtMin, nullptr, N);

  // Sinkhorn EMD: one workgroup per batch, fully LDS-resident
  sinkhorn_kernel<<<B, 1024, 0, stream>>>(pc1_0, pc2, emdSum);

  finalize_kernel<<<1, 256, 0, stream>>>(rowMin1, colMin1, rowMinS, colMinS,
                                         gtMin, emdSum, pc1_3, (float*)d_out,
                                         nAll, nSeed);
}